// Select_12343736009279
// MI455X (gfx1250) — compile-verified
//
#include <hip/hip_runtime.h>
#include <hip/hip_bf16.h>
#include <math.h>

typedef __attribute__((ext_vector_type(2))) float v2f;
typedef __attribute__((ext_vector_type(8))) float v8f;

#define Bn 32
#define Ln 2048
#define Cn 1024
#define Gn 32
#define Kn 20
#define ML (Bn*Ln)   // 65536 rows

// ---------------- wave32 reduction helper ----------------
__device__ inline float wave_red_add(float v) {
  v += __shfl_down(v, 16, 32);
  v += __shfl_down(v, 8, 32);
  v += __shfl_down(v, 4, 32);
  v += __shfl_down(v, 2, 32);
  v += __shfl_down(v, 1, 32);
  return v;
}

// ---------------- CDNA5 async global->LDS copy (ASYNCcnt-tracked) ----------------
// Per CDNA5 ISA: each lane copies 16B from its global address to
// the LDS address in its vdst. No VGPR staging; overlap with compute, then s_wait_asynccnt.
__device__ __forceinline__ void async_copy16(unsigned lds_off, const void* gptr) {
  asm volatile("global_load_async_to_lds_b128 %0, %1, off"
               :
               : "v"(lds_off), "v"((unsigned long long)(size_t)gptr)
               : "memory");
}
__device__ __forceinline__ void wait_async0() {
  asm volatile("s_wait_asynccnt 0" ::: "memory");
}

// ---------------- Kernel 1: Y = X @ W^T + b  (f32 WMMA 16x16x4) ----------------
// X: [65536,1024], W: [1024,1024] row-major (Y[m,d] = sum_c X[m,c]*W[d,c]).
// 128x128 block tile, BK=32, 8 waves x (2x4) 16x16 tiles, double-buffered LDS
// filled with global_load_async_to_lds_b128.
#define BM 128
#define BN 128
#define BK 32
#define LDST 36   // padded LDS row stride (144B = 9*16B: b128-aligned, conflict-free)

__global__ __launch_bounds__(256) void gemm1_wmma(
    const float* __restrict__ X, const float* __restrict__ W,
    const float* __restrict__ bias, float* __restrict__ Y)
{
  __shared__ float lx[2][BM * LDST];
  __shared__ float lw[2][BN * LDST];
  const int t    = threadIdx.x;
  const int lane = t & 31;
  const int wave = t >> 5;
  const int wm   = wave >> 1;        // 0..3 : wave row
  const int wn   = wave & 1;         // 0..1 : wave col
  const int m0   = blockIdx.x * BM;
  const int n0   = blockIdx.y * BN;
  const int lm   = lane & 15;
  const int ko   = (lane >> 4) << 1; // lanes 0-15 -> K{0,1}, 16-31 -> K{2,3}

  // per-thread fill slots: 4 x b128 per tile per buffer
  int row_[4], c4_[4];
  #pragma unroll
  for (int i = 0; i < 4; ++i) {
    int fi = t + i * 256;
    row_[i] = fi >> 3;
    c4_[i]  = (fi & 7) << 2;
  }
  const unsigned lxb[2] = { (unsigned)(size_t)&lx[0][0], (unsigned)(size_t)&lx[1][0] };
  const unsigned lwb[2] = { (unsigned)(size_t)&lw[0][0], (unsigned)(size_t)&lw[1][0] };

  auto issue_tile = [&](int k0, int buf) {
    #pragma unroll
    for (int i = 0; i < 4; ++i) {
      unsigned off = (unsigned)(row_[i] * LDST + c4_[i]) * 4u;
      async_copy16(lxb[buf] + off, X + (size_t)(m0 + row_[i]) * Cn + k0 + c4_[i]);
      async_copy16(lwb[buf] + off, W + (size_t)(n0 + row_[i]) * Cn + k0 + c4_[i]);
    }
  };

  v8f acc[2][4] = {};

  const int abase = (wm * 32 + lm) * LDST + ko;
  const int bbase = (wn * 64 + lm) * LDST + ko;

  issue_tile(0, 0);
  wait_async0();
  __syncthreads();

  for (int k0 = 0; k0 < Cn; k0 += BK) {
    const int cur = (k0 / BK) & 1;
    if (k0 + BK < Cn) issue_tile(k0 + BK, cur ^ 1);   // overlap fill with WMMA

    const float* cx = &lx[cur][0];
    const float* cw = &lw[cur][0];
    #pragma unroll
    for (int kk = 0; kk < BK; kk += 4) {
      v2f a0 = *(const v2f*)&cx[abase + kk];
      v2f a1 = *(const v2f*)&cx[abase + 16 * LDST + kk];
      v2f b0 = *(const v2f*)&cw[bbase + kk];
      v2f b1 = *(const v2f*)&cw[bbase + 16 * LDST + kk];
      v2f b2 = *(const v2f*)&cw[bbase + 32 * LDST + kk];
      v2f b3 = *(const v2f*)&cw[bbase + 48 * LDST + kk];
      acc[0][0] = __builtin_amdgcn_wmma_f32_16x16x4_f32(false, a0, false, b0, (short)0, acc[0][0], false, false);
      acc[0][1] = __builtin_amdgcn_wmma_f32_16x16x4_f32(false, a0, false, b1, (short)0, acc[0][1], false, false);
      acc[0][2] = __builtin_amdgcn_wmma_f32_16x16x4_f32(false, a0, false, b2, (short)0, acc[0][2], false, false);
      acc[0][3] = __builtin_amdgcn_wmma_f32_16x16x4_f32(false, a0, false, b3, (short)0, acc[0][3], false, false);
      acc[1][0] = __builtin_amdgcn_wmma_f32_16x16x4_f32(false, a1, false, b0, (short)0, acc[1][0], false, false);
      acc[1][1] = __builtin_amdgcn_wmma_f32_16x16x4_f32(false, a1, false, b1, (short)0, acc[1][1], false, false);
      acc[1][2] = __builtin_amdgcn_wmma_f32_16x16x4_f32(false, a1, false, b2, (short)0, acc[1][2], false, false);
      acc[1][3] = __builtin_amdgcn_wmma_f32_16x16x4_f32(false, a1, false, b3, (short)0, acc[1][3], false, false);
    }
    wait_async0();      // own async fills of next buffer complete
    __syncthreads();    // all waves' fills complete; all reads of cur complete
  }

  // epilogue: C/D layout -> lanes 0-15: M=v, N=lane; lanes 16-31: M=v+8, N=lane-16
  const int half = (lane >> 4) << 3;
  #pragma unroll
  for (int j = 0; j < 4; ++j) {
    int col = n0 + wn * 64 + j * 16 + lm;
    float bv = bias[col];
    #pragma unroll
    for (int i = 0; i < 2; ++i) {
      int rbase = m0 + wm * 32 + i * 16 + half;
      #pragma unroll
      for (int v = 0; v < 8; ++v) {
        Y[(size_t)(rbase + v) * Cn + col] = acc[i][j][v] + bv;
      }
    }
  }
}

// ---------------- Kernel 2: per-(b,c4) mean/max pooling over L (b128 loads) ----------------
__global__ __launch_bounds__(256) void pool_kernel(
    const float* __restrict__ Y, float* __restrict__ xmean, float* __restrict__ xmax)
{
  int idx = blockIdx.x * 256 + threadIdx.x;   // 8192 threads: (b, c/4)
  int b = idx >> 8;
  int c4 = idx & 255;
  const float4* p = (const float4*)(Y + (size_t)b * Ln * Cn) + c4;
  float4 s  = {0.f, 0.f, 0.f, 0.f};
  float4 mx = {-INFINITY, -INFINITY, -INFINITY, -INFINITY};
  for (int l = 0; l < Ln; ++l) {
    float4 v = p[(size_t)l * (Cn / 4)];
    s.x += v.x; s.y += v.y; s.z += v.z; s.w += v.w;
    mx.x = fmaxf(mx.x, v.x); mx.y = fmaxf(mx.y, v.y);
    mx.z = fmaxf(mx.z, v.z); mx.w = fmaxf(mx.w, v.w);
  }
  const float inv = 1.0f / (float)Ln;
  float4* om = (float4*)(xmean + (size_t)b * Cn) + c4;
  float4* ox = (float4*)(xmax  + (size_t)b * Cn) + c4;
  *om = make_float4(s.x * inv, s.y * inv, s.z * inv, s.w * inv);
  *ox = mx;
}

// ---------------- Kernel 3: per-batch GN + fold W1 into u-vectors ----------------
__global__ __launch_bounds__(256) void gn_prep_kernel(
    const float* __restrict__ xmean, const float* __restrict__ xmax,
    const float* __restrict__ Wm, const float* __restrict__ bm,
    const float* __restrict__ Wa, const float* __restrict__ ba,
    const float* __restrict__ g1w, const float* __restrict__ g1b,
    const float* __restrict__ g2w, const float* __restrict__ g2b,
    const float* __restrict__ W1, const float* __restrict__ b1,
    float* __restrict__ uM, float* __restrict__ uA,
    float* __restrict__ cM, float* __restrict__ cA)
{
  __shared__ float sm[Cn], sa[Cn];
  __shared__ float lM[Cn], lA[Cn];
  __shared__ float gmu1[Gn], gsc1[Gn], gmu2[Gn], gsc2[Gn];
  __shared__ float rbuf[256];
  const int b = blockIdx.x, t = threadIdx.x;
  const int lane = t & 31, wave = t >> 5;
  const int GS = Cn / Gn; // 32

  for (int c = t; c < Cn; c += 256) { sm[c] = xmean[b * Cn + c]; sa[c] = xmax[b * Cn + c]; }
  __syncthreads();

  // linear_max / linear_Aug (wave per output row, lanes over c -> coalesced)
  for (int d = wave; d < Cn; d += 8) {
    float aM = 0.f, aA = 0.f;
    for (int c = lane; c < Cn; c += 32) {
      aM += sm[c] * Wm[(size_t)d * Cn + c];
      aA += sa[c] * Wa[(size_t)d * Cn + c];
    }
    aM = wave_red_add(aM); aA = wave_red_add(aA);
    if (lane == 0) { lM[d] = aM + bm[d]; lA[d] = aA + ba[d]; }
  }
  __syncthreads();

  // group stats (32 groups of 32 channels, population variance)
  if (t < Gn) {
    float s = 0.f, ss = 0.f;
    for (int i = 0; i < GS; ++i) { float v = lM[t * GS + i]; s += v; ss += v * v; }
    float mu = s / GS; float var = ss / GS - mu * mu;
    gmu1[t] = mu; gsc1[t] = rsqrtf(var + 1e-5f);
  } else if (t < 2 * Gn) {
    int g = t - Gn;
    float s = 0.f, ss = 0.f;
    for (int i = 0; i < GS; ++i) { float v = lA[g * GS + i]; s += v; ss += v * v; }
    float mu = s / GS; float var = ss / GS - mu * mu;
    gmu2[g] = mu; gsc2[g] = rsqrtf(var + 1e-5f);
  }
  __syncthreads();

  for (int d = t; d < Cn; d += 256) {
    int g = d / GS;
    lM[d] = (lM[d] - gmu1[g]) * gsc1[g] * g1w[d] + g1b[d];
    lA[d] = (lA[d] - gmu2[g]) * gsc2[g] * g2w[d] + g2b[d];
  }
  __syncthreads();

  // u_M / u_A: fold W1 (coalesced across threads in c; one shared W1 read feeds both)
  for (int c = t; c < Cn; c += 256) {
    float accM = 0.f, accA = 0.f;
    for (int d = 0; d < Cn; ++d) {
      float w = W1[(size_t)d * Cn + c];
      accM += lM[d] * w; accA += lA[d] * w;
    }
    uM[b * Cn + c] = accM; uA[b * Cn + c] = accA;
  }

  float pM = 0.f, pA = 0.f;
  for (int d = t; d < Cn; d += 256) { pM += lM[d] * b1[d]; pA += lA[d] * b1[d]; }
  rbuf[t] = pM; __syncthreads();
  for (int s = 128; s > 0; s >>= 1) { if (t < s) rbuf[t] += rbuf[t + s]; __syncthreads(); }
  if (t == 0) cM[b] = rbuf[0];
  __syncthreads();
  rbuf[t] = pA; __syncthreads();
  for (int s = 128; s > 0; s >>= 1) { if (t < s) rbuf[t] += rbuf[t + s]; __syncthreads(); }
  if (t == 0) cA[b] = rbuf[0];
}

// ---------------- Kernel 4: out1/out2 batched matvec over Y (b128 loads) ----------------
__global__ __launch_bounds__(256) void matvec_kernel(
    const float* __restrict__ Y,
    const float* __restrict__ uM, const float* __restrict__ uA,
    const float* __restrict__ cM, const float* __restrict__ cA,
    float* __restrict__ o1, float* __restrict__ o2)
{
  int lane = threadIdx.x & 31;
  int wave = threadIdx.x >> 5;
  int r = blockIdx.x * 8 + wave;            // global row, < 65536
  int b = r >> 11;
  const float4* yr = (const float4*)(Y + (size_t)r * Cn);
  const float4* um = (const float4*)(uM + (size_t)b * Cn);
  const float4* ua = (const float4*)(uA + (size_t)b * Cn);
  float aM = 0.f, aA = 0.f;
  for (int c = lane; c < Cn / 4; c += 32) {
    float4 v = yr[c];
    float4 m = um[c];
    float4 a = ua[c];
    aM += v.x * m.x + v.y * m.y + v.z * m.z + v.w * m.w;
    aA += v.x * a.x + v.y * a.y + v.z * a.z + v.w * a.w;
  }
  aM = wave_red_add(aM); aA = wave_red_add(aA);
  if (lane == 0) { o1[r] = aM + cM[b]; o2[r] = aA + cA[b]; }
}

// ---------------- Kernel 5: softmax-product + top-20 (jax tie order) ----------------
// A positive uniform scale (1/(S1*S2)) does not change ordering, so only
// exp(o1-m1)*exp(o2-m2) matters for top-k. Ties broken by smallest index.
__global__ __launch_bounds__(256) void softmax_topk_kernel(
    const float* __restrict__ o1, const float* __restrict__ o2, int* __restrict__ topk)
{
  __shared__ float p[Ln];
  __shared__ float rbuf[256];
  __shared__ int   ribuf[256];
  __shared__ float sM1, sM2;
  const int b = blockIdx.x, t = threadIdx.x;
  const float* a = o1 + b * Ln;
  const float* c = o2 + b * Ln;

  float m1 = -INFINITY, m2 = -INFINITY;
  for (int l = t; l < Ln; l += 256) { m1 = fmaxf(m1, a[l]); m2 = fmaxf(m2, c[l]); }
  rbuf[t] = m1; __syncthreads();
  for (int s = 128; s > 0; s >>= 1) { if (t < s) rbuf[t] = fmaxf(rbuf[t], rbuf[t + s]); __syncthreads(); }
  if (t == 0) sM1 = rbuf[0];
  __syncthreads();
  rbuf[t] = m2; __syncthreads();
  for (int s = 128; s > 0; s >>= 1) { if (t < s) rbuf[t] = fmaxf(rbuf[t], rbuf[t + s]); __syncthreads(); }
  if (t == 0) sM2 = rbuf[0];
  __syncthreads();
  float M1 = sM1, M2 = sM2;

  for (int l = t; l < Ln; l += 256) p[l] = expf(a[l] - M1) * expf(c[l] - M2);
  __syncthreads();

  for (int k = 0; k < Kn; ++k) {
    float bv = -INFINITY; int bi = Ln;
    for (int l = t; l < Ln; l += 256) {
      float v = p[l];
      if (v > bv || (v == bv && l < bi)) { bv = v; bi = l; }
    }
    rbuf[t] = bv; ribuf[t] = bi; __syncthreads();
    if (t == 0) {
      float Bv = rbuf[0]; int Bi = ribuf[0];
      for (int i = 1; i < 256; ++i) {
        if (rbuf[i] > Bv || (rbuf[i] == Bv && ribuf[i] < Bi)) { Bv = rbuf[i]; Bi = ribuf[i]; }
      }
      topk[b * Kn + k] = Bi;
      p[Bi] = -INFINITY;
    }
    __syncthreads();
  }
}

// ---------------- Kernel 6: recompute z = y_sel @ W1^T + b1 for selected rows ----------------
__global__ __launch_bounds__(256) void gather_linear1_kernel(
    const float* __restrict__ Y, const float* __restrict__ W1, const float* __restrict__ b1,
    const int* __restrict__ topk, float* __restrict__ out)
{
  extern __shared__ float rows[];   // Kn * Cn floats (80 KB dynamic LDS)
  const int b = blockIdx.x, t = threadIdx.x;
  const int lane = t & 31, wave = t >> 5;

  for (int k = 0; k < Kn; ++k) {
    int l = topk[b * Kn + k];
    const float* yr = Y + ((size_t)b * Ln + l) * Cn;
    for (int c = t; c < Cn; c += 256) rows[k * Cn + c] = yr[c];
  }
  __syncthreads();

  for (int d = wave; d < Cn; d += 8) {
    float acc[Kn];
    #pragma unroll
    for (int k = 0; k < Kn; ++k) acc[k] = 0.f;
    for (int c = lane; c < Cn; c += 32) {
      float w = W1[(size_t)d * Cn + c];
      #pragma unroll
      for (int k = 0; k < Kn; ++k) acc[k] += w * rows[k * Cn + c];
    }
    #pragma unroll
    for (int k = 0; k < Kn; ++k) {
      float s = wave_red_add(acc[k]);
      if (lane == 0) out[((size_t)b * Kn + k) * Cn + d] = s + b1[d];
    }
  }
}

// ---------------- host launcher ----------------
extern "C" void kernel_launch(void* const* d_in, const int* in_sizes, int n_in,
                              void* d_out, int out_size, void* d_ws, size_t ws_size,
                              hipStream_t stream) {
  const float* x   = (const float*)d_in[0];
  const float* W   = (const float*)d_in[1];
  const float* bb  = (const float*)d_in[2];
  const float* W1  = (const float*)d_in[3];
  const float* b1  = (const float*)d_in[4];
  const float* Wm  = (const float*)d_in[5];
  const float* bm  = (const float*)d_in[6];
  const float* Wa  = (const float*)d_in[7];
  const float* ba  = (const float*)d_in[8];
  const float* g1w = (const float*)d_in[9];
  const float* g1b = (const float*)d_in[10];
  const float* g2w = (const float*)d_in[11];
  const float* g2b = (const float*)d_in[12];
  float* out = (float*)d_out;

  char* ws = (char*)d_ws;
  float* Y     = (float*)ws; ws += (size_t)ML * Cn * sizeof(float);   // 256 MB
  float* xmean = (float*)ws; ws += (size_t)Bn * Cn * sizeof(float);
  float* xmax  = (float*)ws; ws += (size_t)Bn * Cn * sizeof(float);
  float* uM    = (float*)ws; ws += (size_t)Bn * Cn * sizeof(float);
  float* uA    = (float*)ws; ws += (size_t)Bn * Cn * sizeof(float);
  float* cM    = (float*)ws; ws += (size_t)Bn * sizeof(float);
  float* cA    = (float*)ws; ws += (size_t)Bn * sizeof(float);
  float* o1    = (float*)ws; ws += (size_t)Bn * Ln * sizeof(float);
  float* o2    = (float*)ws; ws += (size_t)Bn * Ln * sizeof(float);
  int*   tk    = (int*)ws;   ws += (size_t)Bn * Kn * sizeof(int);

  gemm1_wmma<<<dim3(ML / BM, Cn / BN), 256, 0, stream>>>(x, W, bb, Y);
  pool_kernel<<<(Bn * Cn / 4) / 256, 256, 0, stream>>>(Y, xmean, xmax);
  gn_prep_kernel<<<Bn, 256, 0, stream>>>(xmean, xmax, Wm, bm, Wa, ba,
                                         g1w, g1b, g2w, g2b, W1, b1, uM, uA, cM, cA);
  matvec_kernel<<<ML / 8, 256, 0, stream>>>(Y, uM, uA, cM, cA, o1, o2);
  softmax_topk_kernel<<<Bn, 256, 0, stream>>>(o1, o2, tk);
  gather_linear1_kernel<<<Bn, 256, Kn * Cn * sizeof(float), stream>>>(Y, W1, b1, tk, out);
}